// Fusion_28535762715131
// MI455X (gfx1250) — compile-verified
//
#include <hip/hip_runtime.h>

typedef float v2f __attribute__((ext_vector_type(2)));
typedef float v8f __attribute__((ext_vector_type(8)));

#define K_ELEMS 1000000
#define N_COLS  70400
#define SENTINEL -9999999.0f
#define HSTRIDE 17   // 48-row LDS tiles padded to 17 floats/row (bank-conflict-free)

// ---- monotone float <-> uint mapping so we can use native u32 atomic max ----
__device__ __forceinline__ unsigned f2key(float f) {
  unsigned b = __float_as_uint(f);
  return (b & 0x80000000u) ? ~b : (b | 0x80000000u);
}
__device__ __forceinline__ float key2f(unsigned k) {
  unsigned b = (k & 0x80000000u) ? (k & 0x7FFFFFFFu) : ~k;
  return __uint_as_float(b);
}

__global__ void init_kernel(unsigned* __restrict__ ws, float* __restrict__ out) {
  int i = blockIdx.x * blockDim.x + threadIdx.x;
  if (i < N_COLS) ws[i] = f2key(SENTINEL);
  if (i == 0) out[N_COLS] = 1.0f;   // flag output
}

__global__ void finalize_kernel(const unsigned* __restrict__ ws, float* __restrict__ out) {
  int i = blockIdx.x * blockDim.x + threadIdx.x;
  if (i < N_COLS) out[i] = key2f(ws[i]);
}

// ---- WMMA helpers (fp32 matrix pipe: V_WMMA_F32_16X16X4_F32) ----
__device__ __forceinline__ v8f wmma_f32(v2f a, v2f b, v8f c) {
  // 8 args: (neg_a, A, neg_b, B, c_mod, C, reuse_a, reuse_b)
  return __builtin_amdgcn_wmma_f32_16x16x4_f32(false, a, false, b, (short)0, c,
                                               false, false);
}

// A-matrix fragment (16x4 tile of W, zero-padded). Layout per ISA:
// lane l holds row M = l%16; VGPR0 = K {0|2}, VGPR1 = K {1|3} by lane half.
__device__ __forceinline__ v2f load_wfrag(const float* __restrict__ W, int M, int Kd,
                                          int t, int j, int ln, int half) {
  int row = t * 16 + ln;
  int k0  = 4 * j + 2 * half;
  v2f a;
  a.x = (row < M && k0     < Kd) ? W[row * Kd + k0    ] : 0.0f;
  a.y = (row < M && k0 + 1 < Kd) ? W[row * Kd + k0 + 1] : 0.0f;
  return a;
}

__global__ void __launch_bounds__(256)
mlp_scatter_kernel(const float* __restrict__ x,     // (4, K) row-major
                   const int*   __restrict__ tidx,  // (K, 2) int32
                   const float* __restrict__ W1, const float* __restrict__ b1,
                   const float* __restrict__ W2, const float* __restrict__ b2,
                   const float* __restrict__ W3, const float* __restrict__ b3,
                   const float* __restrict__ W4, const float* __restrict__ b4,
                   unsigned* __restrict__ ws) {
  __shared__ float bias1[32];
  __shared__ float bias2[48];
  __shared__ float bias3[48];
  __shared__ float hA[8][48 * HSTRIDE];  // wave-private activation buffers
  __shared__ float hB[8][48 * HSTRIDE];

  const int tid = threadIdx.x;
  if (tid < 32) bias1[tid] = (tid < 18) ? b1[tid] : 0.0f;
  if (tid < 48) bias2[tid] = (tid < 36) ? b2[tid] : 0.0f;
  if (tid < 48) bias3[tid] = (tid < 36) ? b3[tid] : 0.0f;
  __syncthreads();

  const int lane = tid & 31;
  const int ln   = lane & 15;
  const int half = lane >> 4;
  const int wv   = tid >> 5;
  float* bufA = hA[wv];
  float* bufB = hB[wv];

  // ---- preload all padded weight A-fragments into registers (reused per tile) ----
  v2f a1[2], a2[3][5], a3[3][9], a4[9];
#pragma unroll
  for (int t = 0; t < 2; ++t) a1[t] = load_wfrag(W1, 18, 4, t, 0, ln, half);
#pragma unroll
  for (int t = 0; t < 3; ++t)
#pragma unroll
    for (int j = 0; j < 5; ++j) a2[t][j] = load_wfrag(W2, 36, 18, t, j, ln, half);
#pragma unroll
  for (int t = 0; t < 3; ++t)
#pragma unroll
    for (int j = 0; j < 9; ++j) a3[t][j] = load_wfrag(W3, 36, 36, t, j, ln, half);
#pragma unroll
  for (int j = 0; j < 9; ++j) a4[j] = load_wfrag(W4, 1, 36, 0, j, ln, half);
  const float b4v = b4[0];

  const int nTiles     = K_ELEMS / 16;                       // 62500, exact
  const int waveId     = blockIdx.x * (blockDim.x >> 5) + wv;
  const int waveStride = gridDim.x * (blockDim.x >> 5);
  const v8f z = {};

  for (int tile = waveId; tile < nTiles; tile += waveStride) {
    const int col0 = tile * 16;

    // ---- layer 1: B = x tile (4x16), two M-tiles of W1 ----
    v2f bx;
    bx.x = x[(2 * half + 0) * K_ELEMS + col0 + ln];
    bx.y = x[(2 * half + 1) * K_ELEMS + col0 + ln];
#pragma unroll
    for (int t = 0; t < 2; ++t) {
      v8f d = wmma_f32(a1[t], bx, z);
#pragma unroll
      for (int j = 0; j < 8; ++j) {
        int row = t * 16 + j + half * 8;   // C/D layout: VGPR j = rows j / j+8
        bufA[row * HSTRIDE + ln] = fmaxf(d[j] + bias1[row], 0.0f);
      }
    }
    __builtin_amdgcn_wave_barrier();

    // ---- layer 2: K=18 (5 zero-padded k-steps), 3 M-tiles ----
    v8f d2[3];
#pragma unroll
    for (int t = 0; t < 3; ++t) d2[t] = z;
#pragma unroll
    for (int j = 0; j < 5; ++j) {
      int r0 = 4 * j + 2 * half;
      v2f b;
      b.x = bufA[(r0 + 0) * HSTRIDE + ln];
      b.y = bufA[(r0 + 1) * HSTRIDE + ln];
#pragma unroll
      for (int t = 0; t < 3; ++t) d2[t] = wmma_f32(a2[t][j], b, d2[t]);
    }
#pragma unroll
    for (int t = 0; t < 3; ++t)
#pragma unroll
      for (int j = 0; j < 8; ++j) {
        int row = t * 16 + j + half * 8;
        bufB[row * HSTRIDE + ln] = fmaxf(d2[t][j] + bias2[row], 0.0f);
      }
    __builtin_amdgcn_wave_barrier();

    // ---- layer 3: K=36 (9 k-steps), 3 M-tiles ----
    v8f d3[3];
#pragma unroll
    for (int t = 0; t < 3; ++t) d3[t] = z;
#pragma unroll
    for (int j = 0; j < 9; ++j) {
      int r0 = 4 * j + 2 * half;
      v2f b;
      b.x = bufB[(r0 + 0) * HSTRIDE + ln];
      b.y = bufB[(r0 + 1) * HSTRIDE + ln];
#pragma unroll
      for (int t = 0; t < 3; ++t) d3[t] = wmma_f32(a3[t][j], b, d3[t]);
    }
#pragma unroll
    for (int t = 0; t < 3; ++t)
#pragma unroll
      for (int j = 0; j < 8; ++j) {
        int row = t * 16 + j + half * 8;
        bufA[row * HSTRIDE + ln] = fmaxf(d3[t][j] + bias3[row], 0.0f);
      }
    __builtin_amdgcn_wave_barrier();

    // ---- layer 4: 1x36 dot per column (1 M-tile, 9 k-steps) ----
    v8f d4 = z;
#pragma unroll
    for (int j = 0; j < 9; ++j) {
      int r0 = 4 * j + 2 * half;
      v2f b;
      b.x = bufA[(r0 + 0) * HSTRIDE + ln];
      b.y = bufA[(r0 + 1) * HSTRIDE + ln];
      d4 = wmma_f32(a4[j], b, d4);
    }

    // v for column col0+n lives in D VGPR0, lanes 0..15 (M=0, N=lane)
    float v = d4[0] + b4v;
    if (lane < 16) {
      int c = tidx[(col0 + ln) * 2 + 1];   // idx1
      atomicMax(&ws[c], f2key(v));
    }
    __builtin_amdgcn_wave_barrier();
  }
}

extern "C" void kernel_launch(void* const* d_in, const int* in_sizes, int n_in,
                              void* d_out, int out_size, void* d_ws, size_t ws_size,
                              hipStream_t stream) {
  const float* x    = (const float*)d_in[0];
  const int*   tidx = (const int*)d_in[1];
  const float* W1   = (const float*)d_in[2];
  const float* b1   = (const float*)d_in[3];
  const float* W2   = (const float*)d_in[4];
  const float* b2   = (const float*)d_in[5];
  const float* W3   = (const float*)d_in[6];
  const float* b3   = (const float*)d_in[7];
  const float* W4   = (const float*)d_in[8];
  const float* b4   = (const float*)d_in[9];
  float*    out = (float*)d_out;
  unsigned* ws  = (unsigned*)d_ws;

  (void)in_sizes; (void)n_in; (void)out_size; (void)ws_size;

  int initBlocks = (N_COLS + 256) / 256;  // covers N_COLS + flag slot
  init_kernel<<<initBlocks, 256, 0, stream>>>(ws, out);

  mlp_scatter_kernel<<<1024, 256, 0, stream>>>(x, tidx, W1, b1, W2, b2, W3, b3,
                                               W4, b4, ws);

  int finBlocks = (N_COLS + 255) / 256;
  finalize_kernel<<<finBlocks, 256, 0, stream>>>(ws, out);
}